// GraphSAGE_22625887715698
// MI455X (gfx1250) — compile-verified
//
#include <hip/hip_runtime.h>
#include <math.h>

#define N_NODES 100000
#define N_EDGES 1600000
#define DIM_IN  128
#define DIM_H   128
#define DIM_OUT 64

typedef __attribute__((ext_vector_type(2))) float v2f;
typedef __attribute__((ext_vector_type(8))) float v8f;

// ---------------- degree kernels ----------------
__global__ void gcn_deg_init(float* __restrict__ deg) {
    int i = blockIdx.x * blockDim.x + threadIdx.x;
    if (i < N_NODES) deg[i] = 1.0f;          // self-loop weight
}

__global__ void gcn_deg_count(const long long* __restrict__ src, float* __restrict__ deg) {
    int e = blockIdx.x * blockDim.x + threadIdx.x;
    if (e < N_EDGES) atomicAdd(&deg[(int)src[e]], 1.0f);
}

__global__ void gcn_deg_recip(float* __restrict__ deg) {
    int i = blockIdx.x * blockDim.x + threadIdx.x;
    if (i < N_NODES) deg[i] = 1.0f / fmaxf(deg[i], 1.0f);
}

// ---------------- float4 copy (self-loop init: acc = feat) ----------------
__global__ void gcn_copy4(const float* __restrict__ in, float* __restrict__ out, int n4) {
    int i = blockIdx.x * blockDim.x + threadIdx.x;
    if (i < n4) {
        const float4* pi = (const float4*)in;
        float4*       po = (float4*)out;
        po[i] = pi[i];
    }
}

// ---------------- edge scatter-add SpMM: acc[src] += feat[dst] ----------------
// One wave (32 lanes) per edge; each lane moves 4 contiguous floats (128 feats).
__global__ void gcn_edge_spmm(const long long* __restrict__ src,
                              const long long* __restrict__ dst,
                              const float* __restrict__ feat,
                              float* __restrict__ acc) {
    long long g = (long long)blockIdx.x * blockDim.x + threadIdx.x;
    int e    = (int)(g >> 5);
    int lane = (int)(g & 31);
    if (e >= N_EDGES) return;
    int s = (int)src[e];
    int d = (int)dst[e];
    const float4 v = *(const float4*)(feat + (size_t)d * DIM_H + lane * 4);
    float* p = acc + (size_t)s * DIM_H + lane * 4;
    atomicAdd(p + 0, v.x);
    atomicAdd(p + 1, v.y);
    atomicAdd(p + 2, v.z);
    atomicAdd(p + 3, v.w);
}

// ---------------- WMMA GEMM: out = act( (A*rdeg) @ W^T + bias ) ----------------
// A: [N_NODES, 128] (aggregated features), W: [nout, 128] row-major (torch Linear),
// so B[k][n] = W[n][k].  Block = (nout/16) waves; each wave owns one 16-col tile.
// M-tile of 16 nodes staged in LDS with 1/deg scaling fused.
// V_WMMA_F32_16X16X4_F32 fragment layouts per ISA 7.12.2:
//   A 16x4 : vgpr0 = A[m][k0] (lanes0-15) / A[m][k2] (lanes16-31), vgpr1 = k1/k3
//   B 4x16 : vgpr0 = B[k0][n] / B[k2][n],  vgpr1 = B[k1][n] / B[k3][n]
//   C/D    : vgpr r = D[r][n] (lanes0-15) / D[r+8][n] (lanes16-31)
__global__ void gcn_gemm_wmma(const float* __restrict__ A,
                              const float* __restrict__ rdeg,
                              const float* __restrict__ W,
                              const float* __restrict__ bias,
                              float* __restrict__ out,
                              int nout, int doRelu) {
    __shared__ float As[16][DIM_H];   // 8 KB

    const int m0  = blockIdx.x * 16;
    const int tid = threadIdx.x;

    // stage A tile with degree normalization fused
    for (int i = tid; i < 16 * DIM_H; i += blockDim.x) {
        int r = i >> 7, c = i & 127;
        As[r][c] = A[(size_t)(m0 + r) * DIM_H + c] * rdeg[m0 + r];
    }
    __syncthreads();

    const int wave   = tid >> 5;
    const int lane   = tid & 31;
    const int n_base = wave * 16;
    const int lhalf  = (lane < 16) ? 0 : 2;   // K sub-offset for upper half-wave
    const int mn     = lane & 15;             // row index (A) == col index (B)

    v8f c = {};
    const float* wrow = W + (size_t)(n_base + mn) * DIM_H;   // B column n = W row n

    #pragma unroll 8
    for (int kk = 0; kk < DIM_H; kk += 4) {
        v2f a, b;
        a.x = As[mn][kk + lhalf];
        a.y = As[mn][kk + lhalf + 1];
        b.x = wrow[kk + lhalf];
        b.y = wrow[kk + lhalf + 1];
        // 8 args: (neg_a, A, neg_b, B, c_mod, C, reuse_a, reuse_b)
        c = __builtin_amdgcn_wmma_f32_16x16x4_f32(false, a, false, b, (short)0, c,
                                                  false, false);
    }

    const float bv   = bias[n_base + mn];
    const int   moff = (lane < 16) ? 0 : 8;
    #pragma unroll
    for (int r = 0; r < 8; ++r) {
        float v = c[r] + bv;
        if (doRelu) v = fmaxf(v, 0.0f);
        out[(size_t)(m0 + moff + r) * nout + n_base + mn] = v;
    }
}

// ---------------- log_softmax over 64 cols; one wave32 per node ----------------
__global__ void gcn_logsoftmax64(const float* __restrict__ z, float* __restrict__ out) {
    long long g = (long long)blockIdx.x * blockDim.x + threadIdx.x;
    int node = (int)(g >> 5);
    int lane = (int)(g & 31);
    if (node >= N_NODES) return;
    float v0 = z[(size_t)node * DIM_OUT + lane];
    float v1 = z[(size_t)node * DIM_OUT + lane + 32];
    float mx = fmaxf(v0, v1);
    #pragma unroll
    for (int off = 16; off > 0; off >>= 1) mx = fmaxf(mx, __shfl_xor(mx, off, 32));
    float s = expf(v0 - mx) + expf(v1 - mx);
    #pragma unroll
    for (int off = 16; off > 0; off >>= 1) s += __shfl_xor(s, off, 32);
    float lse = mx + logf(s);
    out[(size_t)node * DIM_OUT + lane]      = v0 - lse;
    out[(size_t)node * DIM_OUT + lane + 32] = v1 - lse;
}

// ---------------- launch ----------------
extern "C" void kernel_launch(void* const* d_in, const int* in_sizes, int n_in,
                              void* d_out, int out_size, void* d_ws, size_t ws_size,
                              hipStream_t stream) {
    const float*     x  = (const float*)d_in[0];
    const long long* ei = (const long long*)d_in[1];   // int64 [2, N_EDGES]
    const float*     W1 = (const float*)d_in[2];
    const float*     b1 = (const float*)d_in[3];
    const float*     W2 = (const float*)d_in[4];
    const float*     b2 = (const float*)d_in[5];
    float*           out = (float*)d_out;

    const long long* src = ei;              // row index
    const long long* dst = ei + N_EDGES;    // col index

    // workspace layout (floats): deg | bufA (100K x 128) | bufB (100K x 128) | z (100K x 64)
    float* deg  = (float*)d_ws;
    float* bufA = deg  + N_NODES;
    float* bufB = bufA + (size_t)N_NODES * DIM_H;
    float* zbuf = bufB + (size_t)N_NODES * DIM_H;

    const int nodeBlocks = (N_NODES + 255) / 256;
    const int edgeBlocks = (N_EDGES + 255) / 256;
    const long long spmmThreads = (long long)N_EDGES * 32;
    const int spmmBlocks = (int)((spmmThreads + 255) / 256);
    const int feat4      = N_NODES * DIM_H / 4;
    const int copyBlocks = (feat4 + 255) / 256;
    const int gemmBlocks = N_NODES / 16;           // 6250, exact
    const int lsmBlocks  = (int)(((long long)N_NODES * 32 + 255) / 256);

    // degrees -> 1/deg
    gcn_deg_init <<<nodeBlocks, 256, 0, stream>>>(deg);
    gcn_deg_count<<<edgeBlocks, 256, 0, stream>>>(src, deg);
    gcn_deg_recip<<<nodeBlocks, 256, 0, stream>>>(deg);

    // layer 1: agg = x + scatter(x[dst] -> src);  h1 = relu((agg/deg) @ W1^T + b1)
    gcn_copy4    <<<copyBlocks, 256, 0, stream>>>(x, bufA, feat4);
    gcn_edge_spmm<<<spmmBlocks, 256, 0, stream>>>(src, dst, x, bufA);
    gcn_gemm_wmma<<<gemmBlocks, 256, 0, stream>>>(bufA, deg, W1, b1, bufB, DIM_H, 1);

    // layer 2: agg = h1 + scatter(h1[dst] -> src); z = (agg/deg) @ W2^T + b2
    gcn_copy4    <<<copyBlocks, 256, 0, stream>>>(bufB, bufA, feat4);
    gcn_edge_spmm<<<spmmBlocks, 256, 0, stream>>>(src, dst, bufB, bufA);
    gcn_gemm_wmma<<<gemmBlocks, 128, 0, stream>>>(bufA, deg, W2, b2, zbuf, DIM_OUT, 0);

    // log_softmax
    gcn_logsoftmax64<<<lsmBlocks, 256, 0, stream>>>(zbuf, out);
}